// ESLLossFunction_33457795236321
// MI455X (gfx1250) — compile-verified
//
#include <hip/hip_runtime.h>
#include <stdint.h>

#define NUM_BINS 21
#define TPB 256

typedef float v2f __attribute__((ext_vector_type(2)));
typedef float v4f __attribute__((ext_vector_type(4)));
typedef float v8f __attribute__((ext_vector_type(8)));
typedef unsigned int u32x4 __attribute__((ext_vector_type(4)));
typedef int i32x4 __attribute__((ext_vector_type(4)));
typedef int i32x8 __attribute__((ext_vector_type(8)));

#if defined(__has_builtin)
#if __has_builtin(__builtin_amdgcn_tensor_load_to_lds)
#define HAVE_TDM 1
#endif
#if __has_builtin(__builtin_amdgcn_s_wait_tensorcnt)
#define HAVE_TENSORCNT 1
#endif
#endif

// ---------------------------------------------------------------------------
// JAX Threefry-2x32 (key = jax.random.key(42) -> (0, 42)), 20 rounds.
// ---------------------------------------------------------------------------
__device__ __forceinline__ uint32_t rotl32(uint32_t x, uint32_t n) {
  return (x << n) | (x >> (32u - n));
}

__device__ __forceinline__ void threefry2x32(uint32_t c0, uint32_t c1,
                                             uint32_t& o0, uint32_t& o1) {
  const uint32_t k0 = 0u, k1 = 42u;
  const uint32_t k2 = k0 ^ k1 ^ 0x1BD11BDAu;
  uint32_t x0 = c0 + k0, x1 = c1 + k1;
#define TF_R(r) { x0 += x1; x1 = rotl32(x1, (r)); x1 ^= x0; }
  TF_R(13) TF_R(15) TF_R(26) TF_R(6)   x0 += k1; x1 += k2 + 1u;
  TF_R(17) TF_R(29) TF_R(16) TF_R(24)  x0 += k2; x1 += k0 + 2u;
  TF_R(13) TF_R(15) TF_R(26) TF_R(6)   x0 += k0; x1 += k1 + 3u;
  TF_R(17) TF_R(29) TF_R(16) TF_R(24)  x0 += k1; x1 += k2 + 4u;
  TF_R(13) TF_R(15) TF_R(26) TF_R(6)   x0 += k2; x1 += k0 + 5u;
#undef TF_R
  o0 = x0; o1 = x1;
}

// ---------------------------------------------------------------------------
// Deterministic f32 sum of 256 LDS floats using V_WMMA_F32_16X16X4_F32.
// vals as M[m][k] = base[m*16+k]; 4 chained 16x16x4 WMMAs vs ones-B give
// row sums; fold 8 accumulator VGPRs + one lane-16 shuffle.
// Must be executed by a full wave (EXEC all ones).
// ---------------------------------------------------------------------------
__device__ __forceinline__ float wmma_sum256(const float* base) {
  const int lane = (int)(threadIdx.x & 31u);
  const int row  = lane & 15;
  const int koff = (lane >> 4) << 1;  // lanes 0-15 -> K 0,1 ; lanes 16-31 -> K 2,3
  v2f b = {1.0f, 1.0f};
  v8f acc = {0.f, 0.f, 0.f, 0.f, 0.f, 0.f, 0.f, 0.f};
#pragma unroll
  for (int c = 0; c < 4; ++c) {
    v2f a = { base[row * 16 + 4 * c + koff],
              base[row * 16 + 4 * c + koff + 1] };
    acc = __builtin_amdgcn_wmma_f32_16x16x4_f32(
        /*neg_a=*/false, a, /*neg_b=*/false, b,
        /*c_mod=*/(short)0, acc, /*reuse_a=*/false, /*reuse_b=*/false);
  }
  float s = ((acc[0] + acc[1]) + (acc[2] + acc[3])) +
            ((acc[4] + acc[5]) + (acc[6] + acc[7]));
  s += __shfl_xor(s, 16, 32);
  return s;  // uniform across the wave
}

// ---------------------------------------------------------------------------
// Issue one TDM transfer: contiguous `nfloat` f32 from global -> LDS.
// Descriptor per CDNA5 ISA (1-row 2D tile).
// Call from one full wave only; values are wave-uniform.
// ---------------------------------------------------------------------------
#ifdef HAVE_TDM
__device__ __forceinline__ void tdm_load_tile(const float* gsrc, float* ldsdst,
                                              int nfloat) {
  const uint64_t ga = (uint64_t)(uintptr_t)gsrc;            // byte address
  const uint32_t la = (uint32_t)(uintptr_t)ldsdst;          // LDS byte offset
  u32x4 g0;
  g0[0] = 1u;                                   // count=1, is_restore=0, no gather
  g0[1] = la;                                   // lds_addr
  g0[2] = (uint32_t)(ga & 0xFFFFFFFFu);         // global_addr[31:0]
  g0[3] = (uint32_t)((ga >> 32) & 0x01FFFFFFu)  // global_addr[56:32]
          | (2u << 30);                         // type = 2 ("image")
  const uint32_t nf = (uint32_t)nfloat;
  i32x8 g1;
  g1[0] = (int)(2u << 16);                      // wg_mask=0, data_size=2 (4B)
  g1[1] = (int)((nf & 0xFFFFu) << 16);          // tensor_dim0[15:0]
  g1[2] = (int)(((nf >> 16) & 0xFFFFu)          // tensor_dim0[31:16]
          | (1u << 16));                        // tensor_dim1[15:0] = 1
  g1[3] = (int)((nf & 0xFFFFu) << 16);          // tensor_dim1 hi=0, tile_dim0=nf
  g1[4] = 1;                                    // tile_dim1=1, tile_dim2=0
  g1[5] = (int)nf;                              // tensor_dim0_stride[31:0]
  g1[6] = 0;                                    // stride0 hi, stride1 lo
  g1[7] = 0;                                    // stride1 hi
  i32x4 g2 = {0, 0, 0, 0};                      // dims 2/3 unused (tile_dim3=0)
  i32x4 g3 = {0, 0, 0, 0};                      // dim 4 unused
  i32x8 g4 = {0, 0, 0, 0, 0, 0, 0, 0};          // extended group (unused here)
  // 6-arg form (clang-23 / therock-10.0 toolchain)
  __builtin_amdgcn_tensor_load_to_lds(g0, g1, g2, g3, g4, /*cpol=*/0);
}
#endif

__device__ __forceinline__ void wait_tensorcnt0() {
#ifdef HAVE_TENSORCNT
  __builtin_amdgcn_s_wait_tensorcnt(0);
#else
  asm volatile("s_wait_tensorcnt 0x0" ::: "memory");
#endif
}

// ---------------------------------------------------------------------------
// Main kernel: one sample per thread; TDM-staged logits tile in LDS,
// RNG + scalar loads overlapped with the DMA.
// ---------------------------------------------------------------------------
__global__ __launch_bounds__(TPB) void esl_main(
    const float* __restrict__ logits, const float* __restrict__ exp_s,
    const float* __restrict__ true_s, const float* __restrict__ cw,
    float* __restrict__ kl_out, float* __restrict__ mse_out,
    float* __restrict__ partials, int n, int nblk) {
  __shared__ float lds[TPB * NUM_BINS];  // 21504 B
  const int t = (int)threadIdx.x;
  const int blk = (int)blockIdx.x;
  const long long sbase = (long long)blk * TPB;
  const int ns = min(TPB, n - (int)sbase);
  const int nfloat = ns * NUM_BINS;

#ifdef HAVE_TDM
  if (t < 32) {  // wave 0 issues the tensor DMA (EXEC ignored by TDM)
    tdm_load_tile(logits + sbase * NUM_BINS, lds, nfloat);
  }
#else
  {  // fallback: coalesced b128 non-temporal loads -> LDS
    const v4f* g4 = (const v4f*)(logits + sbase * NUM_BINS);
    v4f* l4 = (v4f*)lds;
    const int nv4 = nfloat >> 2;
    for (int idx = t; idx < nv4; idx += TPB)
      l4[idx] = __builtin_nontemporal_load(g4 + idx);
    for (int idx = (nv4 << 2) + t; idx < nfloat; idx += TPB)
      lds[idx] = logits[sbase * NUM_BINS + idx];
  }
#endif

  // ---- work independent of the logits tile: overlap with the DMA ----
  float sv = 0.f, ev = 0.f, w = 0.f, inv = 0.f;
  if (t < ns) {
    const int i = (int)sbase + t;
    sv = true_s[i];
    ev = exp_s[i];
    w  = cw[i];

    // std jitter: JAX uniform(key(42), [B], -0.05, 0.05) via Threefry-2x32
    const uint32_t un = (uint32_t)n;
    const uint32_t half = (un + (un & 1u)) >> 1;
    uint32_t c0, c1; int second;
    if ((uint32_t)i < half) {
      c0 = (uint32_t)i;
      uint32_t p = (uint32_t)i + half;
      c1 = (p < un) ? p : 0u;  // padded counter when B is odd
      second = 0;
    } else {
      c0 = (uint32_t)i - half; c1 = (uint32_t)i; second = 1;
    }
    uint32_t o0, o1; threefry2x32(c0, c1, o0, o1);
    const uint32_t bits = second ? o1 : o0;
    const float u = __uint_as_float(0x3f800000u | (bits >> 9)) - 1.0f;  // [0,1)
    const float jit = fmaxf(-0.05f, fmaf(u, 0.1f, -0.05f));
    const float stdv = 0.3f + jit;
    inv = 1.0f / (2.0f * stdv * stdv);
  }

#ifdef HAVE_TDM
  if (t < 32) wait_tensorcnt0();  // tile resident before the barrier
#endif
  __syncthreads();

  float wkl = 0.f, wmse = 0.f, wv = 0.f;
  if (t < ns) {
    const int i = (int)sbase + t;
    float x[NUM_BINS];
#pragma unroll
    for (int j = 0; j < NUM_BINS; ++j) x[j] = lds[t * NUM_BINS + j];

    float z[NUM_BINS];
    float zmax = -3.4e38f, xmax = -3.4e38f;
#pragma unroll
    for (int j = 0; j < NUM_BINS; ++j) {
      const float d = 0.5f * (float)j - sv;  // bin centers 0,0.5,...,10
      z[j] = -(d * d) * inv;
      zmax = fmaxf(zmax, z[j]);
      xmax = fmaxf(xmax, x[j]);
    }
    float sez = 0.f, sex = 0.f, az = 0.f, bx = 0.f;
#pragma unroll
    for (int j = 0; j < NUM_BINS; ++j) {
      const float ez = expf(z[j] - zmax);
      sez += ez;
      az = fmaf(ez, z[j], az);
      bx = fmaf(ez, x[j], bx);
      sex += expf(x[j] - xmax);
    }
    const float lse_z = zmax + logf(sez);
    const float lse_x = xmax + logf(sex);
    // kl = sum t*(z - lse_z) - sum t*(x - lse_x), with t = ez/sez, sum t = 1
    const float kl = (az - bx) / sez - lse_z + lse_x;
    const float dd = ev - sv;
    const float mse = dd * dd;

    __builtin_nontemporal_store(kl, kl_out + i);
    __builtin_nontemporal_store(mse, mse_out + i);
    wkl = w * kl; wmse = w * mse; wv = w;
  }

  __syncthreads();  // done reading staged logits; reuse LDS for reduction
  lds[t] = wkl; lds[TPB + t] = wmse; lds[2 * TPB + t] = wv;
  __syncthreads();

  if (t < 32) {  // wave 0 (full EXEC): WMMA f32 reduction, deterministic order
    const float s0 = wmma_sum256(lds);
    const float s1 = wmma_sum256(lds + TPB);
    const float s2 = wmma_sum256(lds + 2 * TPB);
    if (t == 0) {
      partials[blk] = s0;
      partials[nblk + blk] = s1;
      partials[2 * nblk + blk] = s2;
    }
  }
}

// ---------------------------------------------------------------------------
// Final kernel: one block; fixed-order strided accumulation + WMMA reduce.
// ---------------------------------------------------------------------------
__global__ __launch_bounds__(TPB) void esl_final(
    const float* __restrict__ partials, int nblk, float* __restrict__ out) {
  __shared__ float lds[3 * TPB];
  const int t = (int)threadIdx.x;
  float a0 = 0.f, a1 = 0.f, a2 = 0.f;
  for (int b = t; b < nblk; b += TPB) {
    a0 += partials[b];
    a1 += partials[nblk + b];
    a2 += partials[2 * nblk + b];
  }
  lds[t] = a0; lds[TPB + t] = a1; lds[2 * TPB + t] = a2;
  __syncthreads();
  if (t < 32) {
    const float skl  = wmma_sum256(lds);
    const float smse = wmma_sum256(lds + TPB);
    const float sw   = wmma_sum256(lds + 2 * TPB);
    if (t == 0) {
      const float wkl = skl / sw, wmse = smse / sw;
      out[0] = 0.9f * wkl + 0.1f * wmse;
      out[1] = wkl;
      out[2] = wmse;
    }
  }
}

// ---------------------------------------------------------------------------
extern "C" void kernel_launch(void* const* d_in, const int* in_sizes, int n_in,
                              void* d_out, int out_size, void* d_ws,
                              size_t ws_size, hipStream_t stream) {
  const float* logits = (const float*)d_in[0];
  const float* exp_s  = (const float*)d_in[1];
  const float* true_s = (const float*)d_in[2];
  const float* cw     = (const float*)d_in[3];
  const int n = in_sizes[1];                 // B
  const int nblk = (n + TPB - 1) / TPB;      // 15625 for B=4M

  float* out = (float*)d_out;                // [total, wkl, wmse, kl[B], mse[B]]
  float* kl_out = out + 3;
  float* mse_out = out + 3 + n;
  float* partials = (float*)d_ws;            // nblk*3 floats

  esl_main<<<nblk, TPB, 0, stream>>>(logits, exp_s, true_s, cw, kl_out,
                                     mse_out, partials, n, nblk);
  esl_final<<<1, TPB, 0, stream>>>(partials, nblk, out);
}